// NeighborSearch_batch_4329327034821
// MI455X (gfx1250) — compile-verified
//
#include <hip/hip_runtime.h>
#include <hip/hip_bf16.h>

typedef __attribute__((ext_vector_type(2))) float v2f;
typedef __attribute__((ext_vector_type(8))) float v8f;

#define WAVES_PER_BLOCK 8
// n = 9225 -> 577 tiles of 16 = 9232, +16 prefetch-pad tile = 9248 entries.
#define NPAD_MAX 9248

// One wave handles 16 queries of one batch. For each 16-point tile, a single
// V_WMMA_F32_16X16X4_F32 produces the 16x16 tile of squared distances via the
// augmented-K trick:
//   A row  (query M): [qx, qy, |q|^2, 1]
//   B col  (point N): [-2x, -2y, 1, |x|^2]   =>  D[M][N] = |q - x|^2
// Both B-operand halves are precomputed once per block into LDS (CDNA5 has
// 320KB/WGP): sB0 feeds lanes 0-15 (K=0,1), sB1 feeds lanes 16-31 (K=2,3).
// The hot loop is then ds_load_b64 + wmma + ballot-compaction, zero global
// traffic and zero divergent operand setup.
__global__ __launch_bounds__(256) void
radius_search_wmma(const float* __restrict__ data,
                   const float* __restrict__ queries,
                   long long* __restrict__ out_idx,
                   int n, int m, int max_k, float r2)
{
    __shared__ v2f sB0[NPAD_MAX];   // (B[0][N], B[1][N]) = (-2x, -2y)
    __shared__ v2f sB1[NPAD_MAX];   // (B[2][N], B[3][N]) = (1, |x|^2)

    const int lane   = threadIdx.x & 31;
    const int wave   = threadIdx.x >> 5;
    const int waveId = blockIdx.x * WAVES_PER_BLOCK + wave;
    const int qtiles = m >> 4;
    const int b      = waveId / qtiles;      // same for all waves in a block
    const int qtile  = waveId % qtiles;

    const float* __restrict__ qb = queries + (size_t)b * m * 2;
    const float* __restrict__ db = data    + (size_t)b * n * 2;

    const int ntiles = (n + 15) >> 4;
    const int npad   = ntiles * 16;

    // ---- Stage B operands for the whole batch into LDS (once per block) ----
    for (int i = threadIdx.x; i < npad + 16; i += 256) {
        const bool valid = i < n;
        const int  pc    = valid ? i : 0;
        const float x = db[pc * 2 + 0];
        const float y = db[pc * 2 + 1];
        v2f b0, b1;
        b0.x = valid ? -2.0f * x : 0.0f;
        b0.y = valid ? -2.0f * y : 0.0f;
        b1.x = valid ? 1.0f : 0.0f;
        b1.y = valid ? (x * x + y * y) : 3.0e38f;   // OOB -> huge dist^2
        sB0[i] = b0;
        sB1[i] = b1;
    }
    __syncthreads();

    // ---- Build A operand (16x4 f32 -> 2 VGPRs/lane) ----
    const int lq = lane & 15;
    const int q  = qtile * 16 + lq;
    const float qx = qb[q * 2 + 0];
    const float qy = qb[q * 2 + 1];
    v2f a;
    if (lane < 16) { a.x = qx; a.y = qy; }
    else           { a.x = qx * qx + qy * qy; a.y = 1.0f; }

    // Lanes 0-15 read sB0 (banks 0-31), lanes 16-31 read sB1 (banks 32-63):
    // conflict-free ds_load_b64 per tile.
    const v2f* __restrict__ half = (lane < 16) ? sB0 : sB1;

    long long* __restrict__ rowPtr = out_idx + ((size_t)b * m + q) * (size_t)max_k;
    int cursor = 0;

    v2f bb = half[lq];                        // tile 0
    for (int t = 0; t < ntiles; ++t) {
        const v2f bnext = half[(t + 1) * 16 + lq];  // prefetch (padded tile)

        v8f c = {};
        // (neg_a, A, neg_b, B, c_mod, C, reuse_a, reuse_b)
        c = __builtin_amdgcn_wmma_f32_16x16x4_f32(
                false, a, false, bb, (short)0, c, false, false);

        // c[r] @ lane L = dist^2 for row M = r + (L>=16 ? 8 : 0), col N = L&15.
        // ballot(c[r] <= r2): bits[15:0] -> row r, bits[31:16] -> row r+8.
        unsigned rowmask = 0;
#pragma unroll
        for (int r = 0; r < 8; ++r) {
            const unsigned bal = __builtin_amdgcn_ballot_w32(c[r] <= r2);
            rowmask = (lane == r)     ? (bal & 0xFFFFu) : rowmask;
            rowmask = (lane == r + 8) ? (bal >> 16)     : rowmask;
        }

        // Lanes 0-15 own one query row each; append ascending indices.
        while (rowmask) {
            const int j = __builtin_ctz(rowmask);
            rowmask &= rowmask - 1;
            if (cursor < max_k) rowPtr[cursor] = (long long)(t * 16 + j);
            ++cursor;
        }
        bb = bnext;
    }

    if (lane < 16) {
        for (int k = cursor; k < max_k; ++k) rowPtr[k] = -1LL;  // pad
    }
}

// Per batch: recover counts from the -1 padding, then block-wide LDS scan
// (Hillis-Steele over 256 partials, 16 rows/thread) to emit row_splits.
__global__ __launch_bounds__(256) void
row_splits_kernel(const long long* __restrict__ out_idx,
                  long long* __restrict__ splits, int m, int max_k)
{
    const int b   = blockIdx.x;
    const int t   = threadIdx.x;          // 256 threads
    const int per = m >> 8;               // rows per thread (16 for m=4096)

    __shared__ long long partial[256];

    const long long* __restrict__ rows = out_idx + (size_t)b * m * (size_t)max_k;

    int local[16];
    long long s = 0;
    for (int i = 0; i < per && i < 16; ++i) {
        const long long* rp = rows + (size_t)(t * per + i) * (size_t)max_k;
        int cval = 0;
        for (int k = 0; k < max_k; ++k) cval += (rp[k] >= 0) ? 1 : 0;
        local[i] = cval;
        s += cval;
    }
    partial[t] = s;
    __syncthreads();

    // inclusive scan of 256 partials
    for (int off = 1; off < 256; off <<= 1) {
        long long v = (t >= off) ? partial[t - off] : 0;
        __syncthreads();
        partial[t] += v;
        __syncthreads();
    }

    long long run = (t == 0) ? 0 : partial[t - 1];
    long long* __restrict__ sb = splits + (size_t)b * (m + 1);
    if (t == 0) sb[0] = 0;
    for (int i = 0; i < per && i < 16; ++i) {
        run += local[i];
        sb[t * per + i + 1] = run;
    }
}

extern "C" void kernel_launch(void* const* d_in, const int* in_sizes, int n_in,
                              void* d_out, int out_size, void* d_ws, size_t ws_size,
                              hipStream_t stream) {
    (void)n_in; (void)d_ws; (void)ws_size;

    const float* data    = (const float*)d_in[0];   // [B, n, 2] f32
    const float* queries = (const float*)d_in[1];   // [B, m, 2] f32

    const int B = 4;
    const int n = in_sizes[0] / (B * 2);            // 9225
    const int m = in_sizes[1] / (B * 2);            // 4096

    long long* out = (long long*)d_out;             // int64 outputs, concatenated
    const long long splits_elems = (long long)B * (m + 1);
    const int max_k = (int)(((long long)out_size - splits_elems) /
                            ((long long)B * m));
    long long* out_splits = out + (long long)B * m * (long long)max_k;

    const float r2 = 0.03f * 0.03f;

    const int qtiles = m / 16;                      // 256
    const int waves  = B * qtiles;                  // 1024
    dim3 grid(waves / WAVES_PER_BLOCK);             // 128 blocks x 256 threads
    dim3 block(32 * WAVES_PER_BLOCK);

    radius_search_wmma<<<grid, block, 0, stream>>>(data, queries, out,
                                                   n, m, max_k, r2);
    row_splits_kernel<<<dim3(B), dim3(256), 0, stream>>>(out, out_splits,
                                                         m, max_k);
}